// BilinearInteraction_76106820485389
// MI455X (gfx1250) — compile-verified
//
#include <hip/hip_runtime.h>

// Bilinear field interaction (FiBiNET-style): out[b,p,:] = (v_i @ W[p]) * v_j
// fp32 end-to-end, computed with V_WMMA_F32_16X16X4_F32 on gfx1250 (wave32).
//
// v2 change vs round 1: W is staged in LDS in K-pair-interleaved form so each
// WMMA B fragment (W[kb][n], W[kb+1][n]) is a single contiguous 8B ds_load_b64
// -> removes the v_dual_mov_b32 repacking churn seen in the round-1 asm.

typedef __attribute__((ext_vector_type(2))) float v2f;
typedef __attribute__((ext_vector_type(4))) float v4f;
typedef __attribute__((ext_vector_type(8))) float v8f;

#define NFIELDS 32
#define EDIM 64
#define NPAIRS 496           // 32*31/2
#define ROWS_PER_BLOCK 64    // 4 waves x 16 rows
#define THREADS 128

// Paired W layout: element (k, n) -> sWp[((k>>1)*WPSTRIDE + n)*2 + (k&1)].
// Pair-row stride 80: 160 floats per pair-row => lanes 16-31 (koff=2, next
// pair-row) land 32 banks away from lanes 0-15 -> conflict-free b64 reads.
#define WPSTRIDE 80
// A staging stride 68: the 64 words of a wave's ds_load_b64 A-fragment read
// ((row*68 + 4*kk + koff) pattern) cover all 64 banks.
#define ASTRIDE 68

__global__ __launch_bounds__(THREADS)
void bilinear_interact_wmma(const float* __restrict__ emb,  // [B, 32, 64]
                            const float* __restrict__ W,    // [496, 64, 64]
                            float* __restrict__ out)        // [B, 496, 64]
{
    __shared__ __align__(16) float sWp[(EDIM / 2) * WPSTRIDE * 2];   // 20.0 KB
    __shared__ __align__(16) float sA[ROWS_PER_BLOCK * ASTRIDE];     // 17.0 KB

    const int pb = blockIdx.y;               // pair index 0..495
    const int rb = blockIdx.x;               // 64-row block index
    const int tid  = threadIdx.x;
    const int wave = tid >> 5;
    const int lane = tid & 31;
    const int b0   = rb * ROWS_PER_BLOCK;

    // Decode upper-triangular (i, j) from flat pair index (scalar, uniform).
    int fi = 0, rem = pb;
    while (rem >= NFIELDS - 1 - fi) { rem -= NFIELDS - 1 - fi; ++fi; }
    const int fj = fi + 1 + rem;

    // ---- Stage W[pb] (64x64) into LDS, K-pair interleaved ----
    {
        const float* gw = W + (size_t)pb * EDIM * EDIM;
        #pragma unroll
        for (int e = tid; e < EDIM * EDIM / 4; e += THREADS) {
            const int idx = e * 4;
            const int k = idx >> 6, c = idx & 63;        // c % 4 == 0
            const v4f val = *(const v4f*)(gw + idx);      // global_load_b128
            float* dst = &sWp[(((k >> 1) * WPSTRIDE) + c) * 2 + (k & 1)];
            dst[0] = val.x; dst[2] = val.y; dst[4] = val.z; dst[6] = val.w;
        }
    }

    // ---- Stage the 64x64 A tile (v_i rows) into LDS, coalesced ----
    {
        #pragma unroll
        for (int e = tid; e < ROWS_PER_BLOCK * EDIM / 4; e += THREADS) {
            const int idx = e * 4;
            const int r = idx >> 6, c = idx & 63;
            const v4f val = *(const v4f*)(emb + ((size_t)(b0 + r) * NFIELDS + fi) * EDIM + c);
            *(v4f*)(&sA[r * ASTRIDE + c]) = val;
        }
    }
    __syncthreads();

    // ---- Per-wave 16x64 output tile via V_WMMA_F32_16X16X4_F32 ----
    // A 16x4 layout: lanes 0-15 hold M=lane, K=0..1; lanes 16-31 hold K=2..3.
    // B 4x16 layout: lanes 0-15 hold N=lane, K=0 / K=1; lanes 16-31 K=2 / K=3.
    const int wr0  = wave * 16;             // wave's first row inside block
    const int mrow = lane & 15;
    const int koff = (lane >> 4) * 2;       // 0 or 2
    const int ncol = lane & 15;

    // Preload all 16 A fragments (ds_load_2addr_b64 pairs -> 32 VGPRs).
    v2f afrag[16];
    #pragma unroll
    for (int kk = 0; kk < 16; ++kk)
        afrag[kk] = *(const v2f*)(&sA[(wr0 + mrow) * ASTRIDE + kk * 4 + koff]);

    // Per-lane constant base into the paired-W layout (floats):
    //   float idx = 320*kk + 160*(koff>>1) + 32*nt + 2*ncol (+ {0,1})
    const int bbase = ((koff >> 1) * WPSTRIDE + ncol) * 2;

    v8f acc0 = {}, acc1 = {}, acc2 = {}, acc3 = {};
    #pragma unroll
    for (int kk = 0; kk < 16; ++kk) {
        const float* brow = &sWp[bbase + kk * (2 * WPSTRIDE * 2)];
        #pragma unroll
        for (int nt = 0; nt < 4; ++nt) {
            const v2f bfrag = *(const v2f*)(brow + nt * 32);   // single ds_load_b64
            // 8 args: (neg_a, A, neg_b, B, c_mod, C, reuse_a, reuse_b)
            if      (nt == 0) acc0 = __builtin_amdgcn_wmma_f32_16x16x4_f32(false, afrag[kk], false, bfrag, (short)0, acc0, false, false);
            else if (nt == 1) acc1 = __builtin_amdgcn_wmma_f32_16x16x4_f32(false, afrag[kk], false, bfrag, (short)0, acc1, false, false);
            else if (nt == 2) acc2 = __builtin_amdgcn_wmma_f32_16x16x4_f32(false, afrag[kk], false, bfrag, (short)0, acc2, false, false);
            else              acc3 = __builtin_amdgcn_wmma_f32_16x16x4_f32(false, afrag[kk], false, bfrag, (short)0, acc3, false, false);
        }
    }

    // ---- Epilogue: out = acc * v_j, C/D layout: VGPR v -> row v + 8*(lane>=16) ----
    const int mhalf = (lane >> 4) * 8;
    #pragma unroll
    for (int nt = 0; nt < 4; ++nt) {
        const v8f acc = (nt == 0) ? acc0 : (nt == 1) ? acc1 : (nt == 2) ? acc2 : acc3;
        const int c = nt * 16 + (lane & 15);
        #pragma unroll
        for (int v = 0; v < 8; ++v) {
            const int r = b0 + wr0 + mhalf + v;                      // global batch row
            const float vj = emb[((size_t)r * NFIELDS + fj) * EDIM + c];
            out[((size_t)r * NPAIRS + pb) * EDIM + c] = acc[v] * vj;
        }
    }
}

extern "C" void kernel_launch(void* const* d_in, const int* in_sizes, int n_in,
                              void* d_out, int out_size, void* d_ws, size_t ws_size,
                              hipStream_t stream) {
    const float* emb = (const float*)d_in[0];   // [B, 32, 64] fp32
    const float* W   = (const float*)d_in[1];   // [496, 64, 64] fp32
    float* out       = (float*)d_out;           // [B, 496, 64] fp32

    const int B = in_sizes[0] / (NFIELDS * EDIM);   // 2048
    dim3 grid(B / ROWS_PER_BLOCK, NPAIRS);          // (32, 496)
    bilinear_interact_wmma<<<grid, THREADS, 0, stream>>>(emb, W, out);
}